// SharpeLoss_29489245454591
// MI455X (gfx1250) — compile-verified
//
#include <hip/hip_runtime.h>
#include <hip/hip_bf16.h>

// Target builtins only resolve on the device pass; host pass reports 0.
#if defined(__HIP_DEVICE_COMPILE__)
#if !__has_builtin(__builtin_amdgcn_wmma_f32_16x16x4_f32)
#error "missing __builtin_amdgcn_wmma_f32_16x16x4_f32 on device pass"
#endif
#if !__has_builtin(__builtin_amdgcn_rcpf)
#error "missing __builtin_amdgcn_rcpf on device pass"
#endif
#endif

typedef __attribute__((ext_vector_type(2))) float v2f;
typedef __attribute__((ext_vector_type(8))) float v8f;

#define BS   256
#define LSZ  2048
#define NSZ  128
#define EPSV 1e-8f
#define LAMV 0.01f

// Stage 1: one block (8 wave32) per batch. Each wave computes Rp for 16 rows
// per pass via V_WMMA_F32_16X16X4_F32 (diagonal of W x r^T), 16 passes cover
// L=2048 rows. Masked one-pass sum / sum-of-squares -> per-batch stats.
//
// r = (P[l+1]-P[l])*rcp(P[l]); since P in [0.5,1.5] no NaN/Inf can occur, and
// the (r<=2 ? r : 0) select maps a hypothetical NaN to 0 exactly like
// nan_to_num + (r<=2) masking in the reference.
__global__ __launch_bounds__(256, 1) void sharpe_stage1(
    const float* __restrict__ W,     // [BS, LSZ, NSZ]
    const float* __restrict__ P,     // [BS, LSZ+1, NSZ]
    const int*   __restrict__ sz,    // [BS]
    float*       __restrict__ stats) // [BS, 4] = {sharpe_list, sharpe, E, var}
{
    const int b    = blockIdx.x;
    const int tid  = threadIdx.x;
    const int wave = tid >> 5;
    const int lane = tid & 31;
    const int half = lane >> 4;   // 0: k-offsets {0,1}; 1: k-offsets {2,3}
    const int mrow = lane & 15;   // matrix row this lane feeds (A and B^T)

    const int n_valid = sz[b];

    const float* Wb = W + (size_t)b * LSZ * NSZ;
    const float* Pb = P + (size_t)b * (LSZ + 1) * NSZ;

    // Diagonal C[m,m] ownership in the 16x16 f32 C layout:
    //   m=0..7  -> lane m    , VGPR m
    //   m=8..15 -> lane m+16 , VGPR m-8   (i.e. lanes 24..31, VGPR lane-24)
    const bool diag = (lane < 8) || (lane >= 24);
    const int  vsel = (lane < 8) ? lane : (lane - 24);

    float s1 = 0.f, s2 = 0.f;

    for (int pass = 0; pass < 16; ++pass) {
        const int l = pass * 128 + wave * 16 + mrow;      // < 2048 always
        const float* wr = Wb + (size_t)l * NSZ + 2 * half;
        const float* pr = Pb + (size_t)l * NSZ + 2 * half;

        v8f c0 = {0.f, 0.f, 0.f, 0.f, 0.f, 0.f, 0.f, 0.f};
        v8f c1 = {0.f, 0.f, 0.f, 0.f, 0.f, 0.f, 0.f, 0.f};

        #pragma unroll 8
        for (int n0 = 0; n0 < NSZ; n0 += 8) {
            // ---- k-chunk A (n0) -> accumulator c0
            {
                v2f a  = *(const v2f*)(wr + n0);
                v2f pi = *(const v2f*)(pr + n0);
                v2f pj = *(const v2f*)(pr + n0 + NSZ);
                float r0 = (pj.x - pi.x) * __builtin_amdgcn_rcpf(pi.x);
                float r1 = (pj.y - pi.y) * __builtin_amdgcn_rcpf(pi.y);
                r0 = (r0 <= 2.f) ? r0 : 0.f;   // also maps NaN -> 0
                r1 = (r1 <= 2.f) ? r1 : 0.f;
                v2f bm; bm.x = r0; bm.y = r1;
                c0 = __builtin_amdgcn_wmma_f32_16x16x4_f32(
                        false, a, false, bm, (short)0, c0, false, false);
            }
            // ---- k-chunk B (n0+4) -> accumulator c1
            {
                v2f a  = *(const v2f*)(wr + n0 + 4);
                v2f pi = *(const v2f*)(pr + n0 + 4);
                v2f pj = *(const v2f*)(pr + n0 + 4 + NSZ);
                float r0 = (pj.x - pi.x) * __builtin_amdgcn_rcpf(pi.x);
                float r1 = (pj.y - pi.y) * __builtin_amdgcn_rcpf(pi.y);
                r0 = (r0 <= 2.f) ? r0 : 0.f;
                r1 = (r1 <= 2.f) ? r1 : 0.f;
                v2f bm; bm.x = r0; bm.y = r1;
                c1 = __builtin_amdgcn_wmma_f32_16x16x4_f32(
                        false, a, false, bm, (short)0, c1, false, false);
            }
        }

        const v8f c = c0 + c1;

        // Extract this lane's diagonal element.
        float d = c[0];
        d = (vsel == 1) ? c[1] : d;
        d = (vsel == 2) ? c[2] : d;
        d = (vsel == 3) ? c[3] : d;
        d = (vsel == 4) ? c[4] : d;
        d = (vsel == 5) ? c[5] : d;
        d = (vsel == 6) ? c[6] : d;
        d = (vsel == 7) ? c[7] : d;

        if (diag && (l < n_valid)) {
            s1 += d;
            s2 += d * d;
        }
    }

    // Wave32 reduction.
    #pragma unroll
    for (int off = 16; off > 0; off >>= 1) {
        s1 += __shfl_xor(s1, off, 32);
        s2 += __shfl_xor(s2, off, 32);
    }

    __shared__ float redA[8];
    __shared__ float redB[8];
    if (lane == 0) { redA[wave] = s1; redB[wave] = s2; }
    __syncthreads();

    if (tid == 0) {
        float S1 = 0.f, S2 = 0.f;
        #pragma unroll
        for (int w = 0; w < 8; ++w) { S1 += redA[w]; S2 += redB[w]; }
        const float n   = (float)n_valid;
        const float E   = S1 / n;
        // sum((Rp-E)^2 * mask) = S2 - n*E^2  (exact algebraic identity)
        const float var = (S2 - S1 * S1 / n) / (n - 1.f);
        const float den = sqrtf(var + EPSV);
        stats[b * 4 + 0] = E - LAMV / den;  // sharpe_list
        stats[b * 4 + 1] = E / den;         // sharpe
        stats[b * 4 + 2] = E;               // expected_Rp
        stats[b * 4 + 3] = var;             // variance_Rp
    }
}

// Stage 2: single block reduces 256 per-batch stat quadruples -> 4 outputs.
__global__ __launch_bounds__(256, 1) void sharpe_stage2(
    const float* __restrict__ stats, float* __restrict__ out)
{
    const int t    = threadIdx.x;       // 256 threads == BS
    const int wave = t >> 5;
    const int lane = t & 31;

    float a0 = stats[t * 4 + 0];
    float a1 = stats[t * 4 + 1];
    float a2 = stats[t * 4 + 2];
    float a3 = stats[t * 4 + 3];

    #pragma unroll
    for (int off = 16; off > 0; off >>= 1) {
        a0 += __shfl_xor(a0, off, 32);
        a1 += __shfl_xor(a1, off, 32);
        a2 += __shfl_xor(a2, off, 32);
        a3 += __shfl_xor(a3, off, 32);
    }

    __shared__ float red[8][4];
    if (lane == 0) {
        red[wave][0] = a0; red[wave][1] = a1;
        red[wave][2] = a2; red[wave][3] = a3;
    }
    __syncthreads();

    if (t == 0) {
        float s0 = 0.f, s1 = 0.f, s2 = 0.f, s3 = 0.f;
        #pragma unroll
        for (int w = 0; w < 8; ++w) {
            s0 += red[w][0]; s1 += red[w][1];
            s2 += red[w][2]; s3 += red[w][3];
        }
        const float inv = 1.0f / (float)BS;
        out[0] = -s0 * inv;   // -mean(sharpe_list)
        out[1] = -s1 * inv;   // -mean(sharpe)
        out[2] =  s2 * inv;   //  mean(expected_Rp)
        out[3] =  s3 * inv;   //  mean(variance_Rp)
    }
}

extern "C" void kernel_launch(void* const* d_in, const int* in_sizes, int n_in,
                              void* d_out, int out_size, void* d_ws, size_t ws_size,
                              hipStream_t stream) {
    const float* W  = (const float*)d_in[0];   // pred_weights [256,2048,128] f32
    const float* P  = (const float*)d_in[1];   // price_list   [256,2049,128] f32
    const int*   sz = (const int*)d_in[2];     // pred_sz      [256] int
    float*       out   = (float*)d_out;        // 4 scalars
    float*       stats = (float*)d_ws;         // [256,4] scratch

    sharpe_stage1<<<BS, 256, 0, stream>>>(W, P, sz, stats);
    sharpe_stage2<<<1, 256, 0, stream>>>(stats, out);
}